// GraphConv1_29738353557517
// MI455X (gfx1250) — compile-verified
//
#include <hip/hip_runtime.h>

// GraphConv: out = relu(concat(F@W, mean_agg(F, edges)@W))
// N=100000 nodes, E=1.6M edges, IN=OUT=32, fp32.
//
// Roofline: edge gather (205MB, L2-resident feature table) + 51.2M f32
// scatter atomics (L2 atomic units) dominate; GEMMs are 0.41 GFLOP and run
// on the WMMA pipe (v_wmma_f32_16x16x4_f32) fused with mean/relu/concat.

typedef __attribute__((ext_vector_type(2))) float v2f;
typedef __attribute__((ext_vector_type(8))) float v8f;

#define FEAT 32

// Native no-return f32 atomic add (GLOBAL_ATOMIC_ADD_F32, GV addressing).
// Guarantees single-instruction L2 atomic regardless of toolchain fp-atomic
// expansion policy (no CAS loop, no VGPR return, STOREcnt-tracked).
__device__ __forceinline__ void atomic_add_f32_native(float* addr, float val) {
    asm volatile("global_atomic_add_f32 %0, %1, off"
                 :
                 : "v"(addr), "v"(val)
                 : "memory");
}

// ---------------------------------------------------------------- zero scratch
__global__ __launch_bounds__(256) void zero_ws_kernel(float* __restrict__ ws, long n) {
    long i = (long)blockIdx.x * blockDim.x + threadIdx.x;
    long stride = (long)gridDim.x * blockDim.x;
    for (; i < n; i += stride) ws[i] = 0.0f;
}

// ------------------------------------------------------------- edge scatter-add
// One 32-lane wave per edge: lane = feature index -> 128B coalesced gather of
// features[src], 32 coalesced global_atomic_add_f32 into agg[dst], +1 count.
__global__ __launch_bounds__(256) void edge_scatter_kernel(
    const float* __restrict__ feat,
    const int*   __restrict__ edges_dst,
    const int*   __restrict__ edges_src,
    float*       __restrict__ agg,   // [N, 32] running sums
    float*       __restrict__ cnt,   // [N]
    int num_edges) {
    long tid = (long)blockIdx.x * blockDim.x + threadIdx.x;
    int e = (int)(tid >> 5);
    int f = (int)(tid & 31);
    if (e >= num_edges) return;
    int s = edges_src[e];
    int d = edges_dst[e];
    float v = feat[(long)s * FEAT + f];
    atomic_add_f32_native(&agg[(long)d * FEAT + f], v);
    if (f == 0) {
        atomic_add_f32_native(&cnt[d], 1.0f);
    }
}

// -------------------------------------------------- fused transform via WMMA
// One wave handles a 16-node tile. Computes relu(F@W) -> out[:,0:32] and
// relu(mean@W) -> out[:,32:64] using V_WMMA_F32_16X16X4_F32 chains (8 K-steps
// over K=32, 4 independent accumulators).
//
// Operand layouts (CDNA5 ISA 7.12.2, 32-bit, wave32; h = lane>>4, m = lane&15):
//   A (16x4, MxK):  row M=m;  VGPR0 -> K = 2h, VGPR1 -> K = 2h+1
//   B (4x16, KxN):  col N=m;  VGPR i, half h -> K = i + 2h
//   C/D (16x16):    VGPR j -> row M = j + 8h, col N = m
__global__ __launch_bounds__(256) void finalize_wmma_kernel(
    const float* __restrict__ feat,    // [N, 32]
    const float* __restrict__ weight,  // [32, 32] row-major
    const float* __restrict__ agg,     // [N, 32] sums
    const float* __restrict__ cnt,     // [N]
    float*       __restrict__ out,     // [N, 64]
    int num_nodes) {
    int wave = blockIdx.x * (blockDim.x >> 5) + (threadIdx.x >> 5);
    int lane = threadIdx.x & 31;
    int row0 = wave * 16;
    if (row0 >= num_nodes) return;            // uniform per wave: EXEC stays all-1s

    int m = lane & 15;
    int h = lane >> 4;
    int g = row0 + m;
    if (g >= num_nodes) g = num_nodes - 1;    // clamp loads for a partial tail tile

    float c   = cnt[g];
    float inv = 1.0f / fmaxf(c, 1.0f);        // exact div: matches reference numerics

    const float* frow = feat + (long)g * FEAT;
    const float* arow = agg  + (long)g * FEAT;

    v8f accS0 = {}; v8f accS1 = {}; v8f accM0 = {}; v8f accM1 = {};

#pragma unroll
    for (int kb = 0; kb < 8; ++kb) {
        int k = kb * 4 + 2 * h;               // this lane's K pair within the block
        v2f a  = *(const v2f*)(frow + k);     // A = features tile
        v2f am;                               // A = mean-aggregated tile
        am.x = arow[k]     * inv;
        am.y = arow[k + 1] * inv;
        v2f b0, b1;                           // B = weight cols [0:16) and [16:32)
        b0.x = weight[(k    ) * FEAT + m];
        b0.y = weight[(k + 1) * FEAT + m];
        b1.x = weight[(k    ) * FEAT + 16 + m];
        b1.y = weight[(k + 1) * FEAT + 16 + m];

        accS0 = __builtin_amdgcn_wmma_f32_16x16x4_f32(false, a,  false, b0, (short)0, accS0, false, false);
        accS1 = __builtin_amdgcn_wmma_f32_16x16x4_f32(false, a,  false, b1, (short)0, accS1, false, false);
        accM0 = __builtin_amdgcn_wmma_f32_16x16x4_f32(false, am, false, b0, (short)0, accM0, false, false);
        accM1 = __builtin_amdgcn_wmma_f32_16x16x4_f32(false, am, false, b1, (short)0, accM1, false, false);
    }

    // Write relu(concat): row stride 64. Divergence here is fine (post-WMMA).
#pragma unroll
    for (int j = 0; j < 8; ++j) {
        int M = j + 8 * h;
        int row = row0 + M;
        if (row < num_nodes) {
            float* orow = out + (long)row * 64;
            orow[m]      = fmaxf(accS0[j], 0.0f);
            orow[16 + m] = fmaxf(accS1[j], 0.0f);
            orow[32 + m] = fmaxf(accM0[j], 0.0f);
            orow[48 + m] = fmaxf(accM1[j], 0.0f);
        }
    }
}

// ---------------------------------------------------------------------- launch
extern "C" void kernel_launch(void* const* d_in, const int* in_sizes, int n_in,
                              void* d_out, int out_size, void* d_ws, size_t ws_size,
                              hipStream_t stream) {
    const float* feat   = (const float*)d_in[0];
    const float* weight = (const float*)d_in[1];
    const int* edst     = (const int*)d_in[2];
    const int* esrc     = (const int*)d_in[3];
    // d_in[4] is num_nodes on device; derive host-side deterministically:
    int N = in_sizes[0] / FEAT;
    int E = in_sizes[2];

    float* agg = (float*)d_ws;                 // [N, 32]
    float* cnt = agg + (size_t)N * FEAT;       // [N]
    float* out = (float*)d_out;                // [N, 64]

    // 1) zero scratch (agg + cnt are contiguous)
    long zn = (long)N * (FEAT + 1);
    zero_ws_kernel<<<1024, 256, 0, stream>>>(agg, zn);

    // 2) edge scatter: one wave (32 lanes) per edge
    long sthreads = (long)E * 32;
    int sblocks = (int)((sthreads + 255) / 256);
    edge_scatter_kernel<<<sblocks, 256, 0, stream>>>(feat, edst, esrc, agg, cnt, E);

    // 3) fused mean + dual GEMM + relu + concat, one wave per 16-row tile
    int tiles = (N + 15) / 16;
    int fblocks = (tiles + 7) / 8;             // 8 waves / 256-thread block
    finalize_wmma_kernel<<<fblocks, 256, 0, stream>>>(feat, weight, agg, cnt, out, N);
}